// GCN_multi_48704929137271
// MI455X (gfx1250) — compile-verified
//
#include <hip/hip_runtime.h>

typedef float v2f __attribute__((ext_vector_type(2)));
typedef float v8f __attribute__((ext_vector_type(8)));

// ---------------------------------------------------------------- utilities
__global__ void zero_f32(float* __restrict__ p, long n) {
  long i = (long)blockIdx.x * blockDim.x + threadIdx.x;
  if (i < n) p[i] = 0.0f;
}

// out[dst[e]][:] += in[src[e]][:]   one thread per (edge, float4 group)
__global__ void scatter_add4(const float* __restrict__ in, float* __restrict__ out,
                             const int* __restrict__ src, const int* __restrict__ dst,
                             int E, int D) {
  int e = blockIdx.x * blockDim.x + threadIdx.x;
  int q = blockIdx.y;                       // 0 .. D/4-1
  if (e >= E) return;
  int s = src[e], d = dst[e];
  float4 v = *(const float4*)(in + (size_t)s * D + q * 4);
  float* o = out + (size_t)d * D + q * 4;
  __hip_atomic_fetch_add(o + 0, v.x, __ATOMIC_RELAXED, __HIP_MEMORY_SCOPE_AGENT);
  __hip_atomic_fetch_add(o + 1, v.y, __ATOMIC_RELAXED, __HIP_MEMORY_SCOPE_AGENT);
  __hip_atomic_fetch_add(o + 2, v.z, __ATOMIC_RELAXED, __HIP_MEMORY_SCOPE_AGENT);
  __hip_atomic_fetch_add(o + 3, v.w, __ATOMIC_RELAXED, __HIP_MEMORY_SCOPE_AGENT);
}

// pooled[batch[n]][f] += relu(agg[n][f])   (HALF = 64 features)
__global__ void pool_relu64(const float* __restrict__ agg, const int* __restrict__ batch,
                            float* __restrict__ pooled, int N) {
  int n = blockIdx.x * blockDim.y + threadIdx.y;
  int f = threadIdx.x;                      // 0..63
  if (n >= N) return;
  float v = fmaxf(agg[(size_t)n * 64 + f], 0.0f);
  __hip_atomic_fetch_add(pooled + (size_t)batch[n] * 64 + f, v,
                         __ATOMIC_RELAXED, __HIP_MEMORY_SCOPE_AGENT);
}

// ------------------------------------------------- fp32 WMMA GEMM (exact)
// out[N, out_ld][col0 : col0+M] = act(A[N,K] @ W[K,M] + bias),  M = NT*16.
// Block: 8 waves, each wave owns 16 rows x full M (NT v8f accumulators).
// W is staged into LDS in KC-row chunks via GLOBAL_LOAD_ASYNC_TO_LDS_B128
// (ASYNCcnt-tracked DMA, no VGPR round trip), shared by all 8 waves.
// LDS rows padded to M+8 floats: half-wave K-split (lanes16-31 read 2 rows
// further = 2*(M+8) dwords = 16 mod 64 banks) is conflict-free.
// B fragments are fetched with ds_load_2addr_b32 (both K-rows of a fragment
// into one consecutive VGPR pair -> no repacking movs).
// Fragment layouts per CDNA5 ISA:
//   A (16x4 f32): lanes 0-15 -> K=k0,k0+1 ; lanes 16-31 -> K=k0+2,k0+3
//   B (4x16 f32): same K split, N = lane&15
//   C/D (16x16 f32): vgpr r -> row r (lanes 0-15) / row r+8 (lanes 16-31)
#define GEMM_WAVES 8
#define KC 64
template <int NT>
__global__ void gemm_wmma_f32(const float* __restrict__ A, const float* __restrict__ W,
                              const float* __restrict__ bias, float* __restrict__ out,
                              int N, int K, int out_ld, int col0, int relu) {
  constexpr int M  = NT * 16;
  constexpr int RS = M + 8;                       // padded LDS row stride (floats)
  constexpr int NITER = (KC * (M / 4)) / 256;     // async b128 loads per thread/chunk
  __shared__ float wlds[KC * RS];

  const int lane    = threadIdx.x;                // 0..31
  const int wv      = threadIdx.y;                // 0..7
  const int tid     = wv * 32 + lane;
  const int rowBase = (blockIdx.x * GEMM_WAVES + wv) * 16;
  const int halfSel = lane >> 4;                  // 0 | 1
  const int lane15  = lane & 15;

  // LDS byte offset of wlds (addrspace(3) pointers are offsets from LDS base)
  const unsigned lbase =
      (unsigned)(__SIZE_TYPE__)(__attribute__((address_space(3))) float*)wlds;

  v8f c[NT] = {};
  // Row m of C depends only on row m of A: clamp OOB rows (their C rows are
  // never stored), so the A load needs no guard and no EXEC manipulation.
  int arow = rowBase + lane15;
  if (arow > N - 1) arow = N - 1;
  const float* Arow = A + (size_t)arow * K;

  for (int kc = 0; kc < K; kc += KC) {            // K is a multiple of KC here
    __syncthreads();                              // previous chunk fully consumed
#pragma unroll
    for (int j = 0; j < NITER; ++j) {             // exact coverage: no guards
      const int i = tid + j * 256;                // float4 index in chunk
      const int r = i / (M / 4);
      const int q = i % (M / 4);
      const unsigned ldst = lbase + (unsigned)((r * RS + q * 4) * 4);
      const unsigned long long ga =
          (unsigned long long)(const void*)(W + (size_t)(kc + r) * M + q * 4);
      asm volatile("global_load_async_to_lds_b128 %0, %1, off"
                   :: "v"(ldst), "v"(ga) : "memory");
    }
    asm volatile("s_wait_asynccnt 0x0" ::: "memory");
    __syncthreads();                              // chunk visible to all waves

#pragma unroll
    for (int k4 = 0; k4 < KC; k4 += 4) {
      v2f a = *(const v2f*)(Arow + kc + k4 + 2 * halfSel);  // one b64 load
      const unsigned wp =
          lbase + (unsigned)(((k4 + 2 * halfSel) * RS + lane15) * 4);
      v2f b[NT];
#pragma unroll
      for (int t = 0; t < NT; ++t)                // fragment pair in one DS op
        asm volatile("ds_load_2addr_b32 %0, %1 offset0:%2 offset1:%3"
                     : "=v"(b[t])
                     : "v"(wp), "i"(t * 16), "i"(RS + t * 16));
      // one wait pinning all fragments: loads -> wait -> WMMA chain
      if constexpr (NT == 8) {
        asm volatile("s_wait_dscnt 0x0"
                     : "+v"(b[0]), "+v"(b[1]), "+v"(b[2]), "+v"(b[3]),
                       "+v"(b[4]), "+v"(b[5]), "+v"(b[6]), "+v"(b[7]));
      } else {
        asm volatile("s_wait_dscnt 0x0"
                     : "+v"(b[0]), "+v"(b[1]), "+v"(b[2]), "+v"(b[3]));
      }
#pragma unroll
      for (int t = 0; t < NT; ++t)                // back-to-back WMMA chain
        c[t] = __builtin_amdgcn_wmma_f32_16x16x4_f32(
            /*neg_a=*/false, a, /*neg_b=*/false, b[t],
            /*c_mod=*/(short)0, c[t], /*reuse_a=*/false, /*reuse_b=*/false);
    }
  }

#pragma unroll
  for (int t = 0; t < NT; ++t) {
    const int   col = t * 16 + lane15;
    const float bv  = bias ? bias[col] : 0.0f;
#pragma unroll
    for (int r = 0; r < 8; ++r) {
      const int orow = rowBase + r + 8 * halfSel;
      if (orow < N) {
        float v = c[t][r] + bv;
        if (relu) v = fmaxf(v, 0.0f);
        out[(size_t)orow * out_ld + col0 + col] = v;
      }
    }
  }
}

// final head: out[b] = dot(h[b,:K], W[:,0]) + b0
__global__ void fc2_dot(const float* __restrict__ h, const float* __restrict__ W,
                        const float* __restrict__ b0, float* __restrict__ out,
                        int B, int K) {
  int i = blockIdx.x * blockDim.x + threadIdx.x;
  if (i >= B) return;
  float acc = 0.0f;
  for (int k = 0; k < K; ++k) acc += h[(size_t)i * K + k] * W[k];
  out[i] = acc + b0[0];
}

// ---------------------------------------------------------------- driver
extern "C" void kernel_launch(void* const* d_in, const int* in_sizes, int n_in,
                              void* d_out, int out_size, void* d_ws, size_t ws_size,
                              hipStream_t stream) {
  const int FEAT = 64, DIM = 128, HALF = 64, ENS = 3, B = 256;

  const float* chr_x  = (const float*)d_in[0];
  const float* slv_x  = (const float*)d_in[1];
  const int*   chr_ei = (const int*)d_in[2];
  const int*   slv_ei = (const int*)d_in[3];
  const int*   chr_bt = (const int*)d_in[4];
  const int*   slv_bt = (const int*)d_in[5];
  const float* chr_W0 = (const float*)d_in[6];
  const float* chr_W1 = (const float*)d_in[7];
  const float* chr_W2 = (const float*)d_in[8];
  const float* slv_W0 = (const float*)d_in[9];
  const float* slv_W1 = (const float*)d_in[10];
  const float* slv_W2 = (const float*)d_in[11];
  const float* cfc_W  = (const float*)d_in[12];
  const float* cfc_b  = (const float*)d_in[13];
  const float* sfc_W  = (const float*)d_in[14];
  const float* sfc_b  = (const float*)d_in[15];
  const float* fc1_W  = (const float*)d_in[16];
  const float* fc1_b  = (const float*)d_in[17];
  const float* fc2_W  = (const float*)d_in[18];
  const float* fc2_b  = (const float*)d_in[19];

  const int N_CHR = in_sizes[0] / FEAT;
  const int N_SLV = in_sizes[1] / FEAT;
  const int E_CHR = in_sizes[2] / 2;
  const int E_SLV = in_sizes[3] / 2;
  const int Nmax  = (N_CHR > N_SLV) ? N_CHR : N_SLV;

  // workspace carve-up (floats)
  float* ws = (float*)d_ws;
  size_t o = 0;
  float* agg0   = ws + o; o += (size_t)Nmax * FEAT;   // shared S·X per graph type
  float* bufA   = ws + o; o += (size_t)Nmax * DIM;
  float* bufB   = ws + o; o += (size_t)Nmax * DIM;
  float* bufC   = ws + o; o += (size_t)Nmax * HALF;
  float* pooled = ws + o; o += (size_t)2 * ENS * B * HALF;  // chr ens0..2, slv ens0..2
  float* hg     = ws + o; o += (size_t)B * (2 * ENS * DIM); // [B,768]
  float* hgo    = ws + o; o += (size_t)B * DIM;             // [B,128]
  (void)ws_size; (void)n_in; (void)out_size;

  auto cdiv = [](long a, long b) { return (unsigned)((a + b - 1) / b); };
  auto zero = [&](float* p, long n) {
    zero_f32<<<cdiv(n, 256), 256, 0, stream>>>(p, n);
  };
  auto scatter = [&](const float* in, float* out, const int* src, const int* dst,
                     int E, int D) {
    dim3 grd(cdiv(E, 256), D / 4);
    scatter_add4<<<grd, 256, 0, stream>>>(in, out, src, dst, E, D);
  };
  auto gemm = [&](const float* A, const float* W, const float* bias, float* out,
                  int N, int K, int M, int out_ld, int col0, int relu) {
    dim3 blk(32, GEMM_WAVES);
    dim3 grd(cdiv(N, 16 * GEMM_WAVES));
    if (M == 128)
      gemm_wmma_f32<8><<<grd, blk, 0, stream>>>(A, W, bias, out, N, K, out_ld, col0, relu);
    else
      gemm_wmma_f32<4><<<grd, blk, 0, stream>>>(A, W, bias, out, N, K, out_ld, col0, relu);
  };

  zero(pooled, (long)2 * ENS * B * HALF);

  auto run_branch = [&](const float* x, const int* ei, const int* batch, int N, int E,
                        const float* W0, const float* W1, const float* W2,
                        float* pooled_base) {
    const int* src = ei;
    const int* dst = ei + E;
    // layer-0 aggregation shared across the ensemble: agg0 = S · X
    zero(agg0, (long)N * FEAT);
    scatter(x, agg0, src, dst, E, FEAT);
    for (int e = 0; e < ENS; ++e) {
      // h1 = relu(agg0 @ W0[e])
      gemm(agg0, W0 + (size_t)e * FEAT * DIM, nullptr, bufA, N, FEAT, DIM, DIM, 0, 1);
      // agg1 = S · h1
      zero(bufB, (long)N * DIM);
      scatter(bufA, bufB, src, dst, E, DIM);
      // h2 = relu(agg1 @ W1[e])
      gemm(bufB, W1 + (size_t)e * DIM * DIM, nullptr, bufA, N, DIM, DIM, DIM, 0, 1);
      // t2 = h2 @ W2[e]  (transform first -> 64-wide scatter)
      gemm(bufA, W2 + (size_t)e * DIM * HALF, nullptr, bufB, N, DIM, HALF, HALF, 0, 0);
      // agg2 = S · t2
      zero(bufC, (long)N * HALF);
      scatter(bufB, bufC, src, dst, E, HALF);
      // pooled[e] = segment_sum_batch(relu(agg2))
      dim3 pblk(64, 4);
      pool_relu64<<<cdiv(N, 4), pblk, 0, stream>>>(bufC, batch,
                                                   pooled_base + (size_t)e * B * HALF, N);
    }
  };

  run_branch(chr_x, chr_ei, chr_bt, N_CHR, E_CHR, chr_W0, chr_W1, chr_W2, pooled);
  run_branch(slv_x, slv_ei, slv_bt, N_SLV, E_SLV, slv_W0, slv_W1, slv_W2,
             pooled + (size_t)ENS * B * HALF);

  // per-ensemble projection heads, written straight into concatenated hg[B,768]
  for (int e = 0; e < ENS; ++e)
    gemm(pooled + (size_t)e * B * HALF, cfc_W + (size_t)e * HALF * DIM,
         cfc_b + (size_t)e * DIM, hg, B, HALF, DIM, 2 * ENS * DIM, e * DIM, 1);
  for (int e = 0; e < ENS; ++e)
    gemm(pooled + (size_t)(ENS + e) * B * HALF, sfc_W + (size_t)e * HALF * DIM,
         sfc_b + (size_t)e * DIM, hg, B, HALF, DIM, 2 * ENS * DIM, (ENS + e) * DIM, 1);

  // fc1: [B,768] @ [768,128] + b, relu
  gemm(hg, fc1_W, fc1_b, hgo, B, 2 * ENS * DIM, DIM, DIM, 0, 1);
  // fc2: [B,128] @ [128,1] + b
  fc2_dot<<<1, 256, 0, stream>>>(hgo, fc2_W, fc2_b, (float*)d_out, B, DIM);
}